// GraphClassifier_86801289052375
// MI455X (gfx1250) — compile-verified
//
#include <hip/hip_runtime.h>
#include <hip/hip_bf16.h>

// ---- problem constants (match reference setup_inputs) ----
#define N_NODES  100000
#define BATCH    256
#define NUM_RELS 200
#define VEC      300
#define DIM      32
#define MODES    6

typedef __attribute__((ext_vector_type(2))) float v2f;
typedef __attribute__((ext_vector_type(8))) float v8f;

// ---------------------------------------------------------------------------
// Kernel 1: scatter batch indices into node->batch lookup tables (pre-filled -1)
// ---------------------------------------------------------------------------
__global__ void scatter_ids_kernel(const int* __restrict__ head_ids,
                                   const int* __restrict__ tail_ids,
                                   int* __restrict__ head_batch,
                                   int* __restrict__ tail_batch) {
    int i = blockIdx.x * blockDim.x + threadIdx.x;
    if (i < BATCH) {
        head_batch[head_ids[i]] = i;
        tail_batch[tail_ids[i]] = i;
    }
}

// ---------------------------------------------------------------------------
// Kernel 2: per-relation embeddings.
//   Rbase[r] = (rel_vectors[r] @ W1 + b1) @ W2 + b2               [200,32]
//   Yrel[m][r] = Rbase[r] @ reld_W[m] + reld_b[m]                 [6,200,32]
// One wave (32 threads) per relation; thread d owns output dim d.
// ---------------------------------------------------------------------------
__global__ __launch_bounds__(32) void rel_emb_kernel(
    const float* __restrict__ rel_vectors, const float* __restrict__ W1,
    const float* __restrict__ b1, const float* __restrict__ W2,
    const float* __restrict__ b2, const float* __restrict__ reld_W,
    const float* __restrict__ reld_b, float* __restrict__ Rbase,
    float* __restrict__ Yrel) {
    __shared__ float t1[DIM];
    __shared__ float rb[DIM];
    const int r = blockIdx.x;
    const int d = threadIdx.x;           // 0..31
    const float* rv = rel_vectors + (size_t)r * VEC;

    float acc = b1[d];
    for (int v = 0; v < VEC; ++v) acc = fmaf(rv[v], W1[v * DIM + d], acc);
    t1[d] = acc;
    __syncthreads();

    acc = b2[d];
    for (int k = 0; k < DIM; ++k) acc = fmaf(t1[k], W2[k * DIM + d], acc);
    rb[d] = acc;
    Rbase[r * DIM + d] = acc;
    __syncthreads();

    for (int m = 0; m < MODES; ++m) {
        float y = reld_b[m * DIM + d];
        const float* Wm = reld_W + (size_t)m * DIM * DIM;
        for (int k = 0; k < DIM; ++k) y = fmaf(rb[k], Wm[k * DIM + d], y);
        Yrel[((size_t)m * NUM_RELS + r) * DIM + d] = y;
    }
}

// ---------------------------------------------------------------------------
// Kernel 3: edge scan -> histogram cnt[mode][batch][rel] and cntTot[mode][batch]
// Memory-bound; fast early-out for edges touching no head/tail node.
// ---------------------------------------------------------------------------
__global__ void edge_scan_kernel(const int* __restrict__ esrc,
                                 const int* __restrict__ edst,
                                 const int* __restrict__ etype,
                                 const int* __restrict__ head_batch,
                                 const int* __restrict__ tail_batch,
                                 unsigned* __restrict__ cnt,
                                 unsigned* __restrict__ cntTot, int nEdges) {
    int e = blockIdx.x * blockDim.x + threadIdx.x;
    if (e >= nEdges) return;
    int s = esrc[e], d = edst[e];
    int hb_s = head_batch[s], hb_d = head_batch[d];
    int tb_s = tail_batch[s], tb_d = tail_batch[d];
    // all four == -1  <=>  AND of them is negative (only negative value is -1)
    if ((hb_s & hb_d & tb_s & tb_d) < 0) return;
    int t = etype[e];
    bool m5 = (hb_s >= 0) && (hb_s == tb_d);
    bool m6 = (hb_d >= 0) && (hb_d == tb_s);
    int seg[MODES];
    seg[0] = ((hb_d >= 0) && !m6) ? hb_d : -1;  // in_edge_out  - mode6
    seg[1] = ((hb_s >= 0) && !m5) ? hb_s : -1;  // out_edge_out - mode5
    seg[2] = ((tb_d >= 0) && !m5) ? tb_d : -1;  // in_edge_in   - mode5
    seg[3] = ((tb_s >= 0) && !m6) ? tb_s : -1;  // out_edge_in  - mode6
    seg[4] = m5 ? hb_s : -1;                    // mode5
    seg[5] = m6 ? hb_d : -1;                    // mode6
#pragma unroll
    for (int m = 0; m < MODES; ++m) {
        int b = seg[m];
        if (b >= 0) {
            atomicAdd(&cnt[((size_t)m * BATCH + b) * NUM_RELS + t], 1u);
            atomicAdd(&cntTot[m * BATCH + b], 1u);
        }
    }
}

// ---------------------------------------------------------------------------
// Kernel 4: WMMA aggregation.
//   agg[m] = cnt[m] @ Yrel[m]   ([256,200] x [200,32]) via V_WMMA_F32_16X16X4_F32
//   rel_neighbor[b] += agg[m][b] / (cntTot[m][b] + 1e-30) / MODES
// One wave per 16x16 output tile. grid = MODES * (BATCH/16) * (DIM/16) = 192.
// A layout (16x4 f32): lanes 0-15 -> M=lane, {v0,v1}={K0,K1}; lanes 16-31 -> {K2,K3}.
// B layout (4x16 f32): N = lane%16; lanes 0-15 -> {K0,K1}; lanes 16-31 -> {K2,K3}.
// D layout: VGPR r -> row r (lanes 0-15) / row r+8 (lanes 16-31), col = lane%16.
// ---------------------------------------------------------------------------
__global__ __launch_bounds__(32) void agg_wmma_kernel(
    const unsigned* __restrict__ cnt, const float* __restrict__ Yrel,
    const unsigned* __restrict__ cntTot, float* __restrict__ rel_neighbor) {
    const int ntiles_n = DIM / 16;    // 2
    const int ntiles_m = BATCH / 16;  // 16
    int tile = blockIdx.x;
    int nt = tile % ntiles_n;
    int mt = (tile / ntiles_n) % ntiles_m;
    int mode = tile / (ntiles_n * ntiles_m);

    const int lane = threadIdx.x;
    const int half = lane >> 4;   // 0 | 1  -> K pair offset 0 | 2
    const int l16 = lane & 15;

    const unsigned* Arow = cnt + ((size_t)mode * BATCH + mt * 16 + l16) * NUM_RELS;
    const float* Bm = Yrel + (size_t)mode * NUM_RELS * DIM + nt * 16 + l16;

    v8f c = {};
    for (int k = 0; k < NUM_RELS; k += 4) {
        int kk = k + half * 2;
        v2f a, b;
        a.x = (float)Arow[kk];
        a.y = (float)Arow[kk + 1];
        b.x = Bm[(size_t)kk * DIM];
        b.y = Bm[(size_t)(kk + 1) * DIM];
        c = __builtin_amdgcn_wmma_f32_16x16x4_f32(
            /*neg_a=*/false, a, /*neg_b=*/false, b,
            /*c_mod=*/(short)0, c, /*reuse_a=*/false, /*reuse_b=*/false);
    }

#pragma unroll
    for (int r = 0; r < 8; ++r) {
        int row = mt * 16 + half * 8 + r;
        int col = nt * 16 + l16;
        float tot = (float)cntTot[mode * BATCH + row];
        float scale = 1.0f / ((tot + 1e-30f) * (float)MODES);
        atomicAdd(&rel_neighbor[row * DIM + col], c[r] * scale);
    }
}

// ---------------------------------------------------------------------------
// Kernel 5: head.  h = relu(concat(rel_neighbor, Rbase[rel_label]) @ conc_W + conc_b)
//                  out = normalize(h) @ fc_W + fc_b
// One wave per batch row.
// ---------------------------------------------------------------------------
__global__ __launch_bounds__(32) void head_kernel(
    const float* __restrict__ rel_neighbor, const float* __restrict__ Rbase,
    const int* __restrict__ rel_labels, const float* __restrict__ conc_W,
    const float* __restrict__ conc_b, const float* __restrict__ fc_W,
    const float* __restrict__ fc_b, float* __restrict__ out) {
    __shared__ float hcat[2 * DIM];
    __shared__ float h[DIM];
    int b = blockIdx.x, k = threadIdx.x;
    hcat[k] = rel_neighbor[b * DIM + k];
    hcat[DIM + k] = Rbase[rel_labels[b] * DIM + k];
    __syncthreads();
    float acc = conc_b[k];
    for (int j = 0; j < 2 * DIM; ++j) acc = fmaf(hcat[j], conc_W[j * DIM + k], acc);
    h[k] = fmaxf(acc, 0.0f);
    __syncthreads();
    if (k == 0) {
        float ss = 0.0f;
        for (int j = 0; j < DIM; ++j) ss = fmaf(h[j], h[j], ss);
        float nrm = fmaxf(sqrtf(ss), 1e-12f);
        float dot = 0.0f;
        for (int j = 0; j < DIM; ++j) dot = fmaf(h[j] / nrm, fc_W[j], dot);
        out[b] = dot + fc_b[0];
    }
}

// ---------------------------------------------------------------------------
extern "C" void kernel_launch(void* const* d_in, const int* in_sizes, int n_in,
                              void* d_out, int out_size, void* d_ws, size_t ws_size,
                              hipStream_t stream) {
    const int* edge_src    = (const int*)d_in[0];
    const int* edge_dst    = (const int*)d_in[1];
    const int* edge_type   = (const int*)d_in[2];
    const int* head_ids    = (const int*)d_in[3];
    const int* tail_ids    = (const int*)d_in[4];
    const int* rel_labels  = (const int*)d_in[5];
    const float* rel_vecs  = (const float*)d_in[6];
    const float* W1        = (const float*)d_in[7];
    const float* b1        = (const float*)d_in[8];
    const float* W2        = (const float*)d_in[9];
    const float* b2        = (const float*)d_in[10];
    const float* reld_W    = (const float*)d_in[11];
    const float* reld_b    = (const float*)d_in[12];
    const float* conc_W    = (const float*)d_in[13];
    const float* conc_b    = (const float*)d_in[14];
    const float* fc_W      = (const float*)d_in[15];
    const float* fc_b      = (const float*)d_in[16];
    float* out             = (float*)d_out;
    const int nEdges       = in_sizes[0];

    // ---- workspace carve-out (256B aligned) ----
    char* base = (char*)d_ws;
    size_t off = 0;
    auto carve = [&](size_t bytes) -> char* {
        char* p = base + off;
        off = (off + bytes + 255) & ~(size_t)255;
        return p;
    };
    int*      head_batch   = (int*)carve((size_t)N_NODES * 4);
    int*      tail_batch   = (int*)carve((size_t)N_NODES * 4);
    unsigned* cnt          = (unsigned*)carve((size_t)MODES * BATCH * NUM_RELS * 4);
    unsigned* cntTot       = (unsigned*)carve((size_t)MODES * BATCH * 4);
    float*    Rbase        = (float*)carve((size_t)NUM_RELS * DIM * 4);
    float*    Yrel         = (float*)carve((size_t)MODES * NUM_RELS * DIM * 4);
    float*    rel_neighbor = (float*)carve((size_t)BATCH * DIM * 4);
    (void)ws_size;

    // ---- per-call re-init (graph-capture-safe stream memsets) ----
    hipMemsetAsync(head_batch, 0xFF, (size_t)N_NODES * 4, stream);             // -1
    hipMemsetAsync(tail_batch, 0xFF, (size_t)N_NODES * 4, stream);             // -1
    hipMemsetAsync(cnt, 0, (size_t)MODES * BATCH * NUM_RELS * 4, stream);
    hipMemsetAsync(cntTot, 0, (size_t)MODES * BATCH * 4, stream);
    hipMemsetAsync(rel_neighbor, 0, (size_t)BATCH * DIM * 4, stream);

    scatter_ids_kernel<<<(BATCH + 255) / 256, 256, 0, stream>>>(
        head_ids, tail_ids, head_batch, tail_batch);

    rel_emb_kernel<<<NUM_RELS, 32, 0, stream>>>(
        rel_vecs, W1, b1, W2, b2, reld_W, reld_b, Rbase, Yrel);

    edge_scan_kernel<<<(nEdges + 255) / 256, 256, 0, stream>>>(
        edge_src, edge_dst, edge_type, head_batch, tail_batch, cnt, cntTot, nEdges);

    agg_wmma_kernel<<<MODES * (BATCH / 16) * (DIM / 16), 32, 0, stream>>>(
        cnt, Yrel, cntTot, rel_neighbor);

    head_kernel<<<BATCH, 32, 0, stream>>>(
        rel_neighbor, Rbase, rel_labels, conc_W, conc_b, fc_W, fc_b, out);
}